// DINOLoss_44083544326419
// MI455X (gfx1250) — compile-verified
//
#include <hip/hip_runtime.h>
#include <math.h>

#define OUT_DIM   65536
#define NCROPS    10
#define BATCH     256
#define N_CENTERS 51

typedef float v2f __attribute__((ext_vector_type(2)));
typedef float v8f __attribute__((ext_vector_type(8)));

__device__ __forceinline__ float teacher_temp_f(int epoch) {
    // np.linspace(0.04, 0.07, 30)[e] for e<30 else 0.07
    if (epoch < 30) return 0.04f + (0.07f - 0.04f) * ((float)epoch / 29.0f);
    return 0.07f;
}

// ---------------------------------------------------------------- init
__global__ void k_init(float* __restrict__ ws) {
    if (threadIdx.x < 64) ws[threadIdx.x] = 0.0f;   // pdots[32] + Lsum[16] + pad
}

// ------------------------------------------------- teacher row stats
// per row r (0..511): x = (t[r,d] - center[targets[r%256],d]) / temp
// store rowmax and 1/sum(exp(x - max))
__global__ void k_teacher_stats(const float* __restrict__ t,
                                const float* __restrict__ center,
                                const int*   __restrict__ targets,
                                const int*   __restrict__ epoch,
                                float* __restrict__ tmax,
                                float* __restrict__ tinvz) {
    __shared__ float red[256];
    const int r = blockIdx.x;                 // 0..511
    const int b = r & (BATCH - 1);
    const float invT = 1.0f / teacher_temp_f(*epoch);
    const float* trow = t + (size_t)r * OUT_DIM;
    const float* crow = center + (size_t)targets[b] * OUT_DIM;

    float m = -1e30f;
    for (int d = threadIdx.x; d < OUT_DIM; d += 256)
        m = fmaxf(m, (trow[d] - crow[d]) * invT);
    red[threadIdx.x] = m; __syncthreads();
    for (int s = 128; s > 0; s >>= 1) {
        if (threadIdx.x < s) red[threadIdx.x] = fmaxf(red[threadIdx.x], red[threadIdx.x + s]);
        __syncthreads();
    }
    m = red[0]; __syncthreads();

    float sum = 0.0f;
    for (int d = threadIdx.x; d < OUT_DIM; d += 256)
        sum += __expf((trow[d] - crow[d]) * invT - m);
    red[threadIdx.x] = sum; __syncthreads();
    for (int s = 128; s > 0; s >>= 1) {
        if (threadIdx.x < s) red[threadIdx.x] += red[threadIdx.x + s];
        __syncthreads();
    }
    if (threadIdx.x == 0) { tmax[r] = m; tinvz[r] = 1.0f / red[0]; }
}

// ------------------------------------------------- student row LSE
// per row r (0..2559): lse = max(s*10) + log(sum exp(s*10 - max));
// accumulate Lsum[v] = sum_b lse[v*B+b]
__global__ void k_student_lse(const float* __restrict__ s,
                              float* __restrict__ Lsum) {
    __shared__ float red[256];
    const int r = blockIdx.x;                 // 0..2559
    const float* srow = s + (size_t)r * OUT_DIM;

    float m = -1e30f;
    for (int d = threadIdx.x; d < OUT_DIM; d += 256)
        m = fmaxf(m, srow[d] * 10.0f);
    red[threadIdx.x] = m; __syncthreads();
    for (int st = 128; st > 0; st >>= 1) {
        if (threadIdx.x < st) red[threadIdx.x] = fmaxf(red[threadIdx.x], red[threadIdx.x + st]);
        __syncthreads();
    }
    m = red[0]; __syncthreads();

    float sum = 0.0f;
    for (int d = threadIdx.x; d < OUT_DIM; d += 256)
        sum += __expf(srow[d] * 10.0f - m);
    red[threadIdx.x] = sum; __syncthreads();
    for (int st = 128; st > 0; st >>= 1) {
        if (threadIdx.x < st) red[threadIdx.x] += red[threadIdx.x + st];
        __syncthreads();
    }
    if (threadIdx.x == 0) atomicAdd(&Lsum[r / BATCH], m + __logf(red[0]));
}

// ------------------------------------------------- WMMA contraction
// dots[i][v] = sum_{b,d} q[i,b,d] * s_raw[v,b,d]  as a 2x10 GEMM with K=B*D.
// One wave = one 16x16x4 f32 WMMA stream over a contiguous K range of one b.
// Unrolled x4 so loads for 4 WMMAs are in flight; student stream is
// non-temporal (read-once) + prefetched ~8KB ahead.
__global__ void k_dots_wmma(const float* __restrict__ s,
                            const float* __restrict__ t,
                            const float* __restrict__ center,
                            const int*   __restrict__ targets,
                            const int*   __restrict__ epoch,
                            const float* __restrict__ tmax,
                            const float* __restrict__ tinvz,
                            float* __restrict__ pdots) {
    const int BLOCKS_PER_B = 16;              // 4 waves/block -> 64 waves per b
    const int b    = blockIdx.x / BLOCKS_PER_B;
    const int blk  = blockIdx.x % BLOCKS_PER_B;
    const int wave = threadIdx.x >> 5;
    const int lane = threadIdx.x & 31;
    const int row  = lane & 15;
    const int half = lane >> 4;

    const int SPAN  = OUT_DIM / 64;           // 1024 K-elements per wave
    const int kbase = (blk * 4 + wave) * SPAN;

    const int iC = (row < 2) ? row : 0;                 // A rows 2..15 zeroed
    const int vC = (row < NCROPS) ? row : (NCROPS - 1); // B cols 10..15 clamped (ignored)
    const float invT = 1.0f / teacher_temp_f(*epoch);

    const float* trow = t + (size_t)(iC * BATCH + b) * OUT_DIM;
    const float* crow = center + (size_t)targets[b] * OUT_DIM;
    const float* srow = s + (size_t)(vC * BATCH + b) * OUT_DIM;
    const float tm = tmax[iC * BATCH + b];
    const float tz = tinvz[iC * BATCH + b];
    const bool qon = (row < 2);
    const int off = 2 * half;                 // lane owns K = k + 2*half + {0,1}

    v8f acc = {0.f, 0.f, 0.f, 0.f, 0.f, 0.f, 0.f, 0.f};

    for (int k = kbase; k < kbase + SPAN; k += 16) {
        // keep the read-once student stream ahead of us
        __builtin_prefetch(srow + k + off + 2048, 0, 1);
#pragma unroll
        for (int kk = 0; kk < 16; kk += 4) {
            const int d0 = k + kk + off;
            v2f tv = *reinterpret_cast<const v2f*>(trow + d0);
            v2f cv = *reinterpret_cast<const v2f*>(crow + d0);
            v2f sv = __builtin_nontemporal_load(reinterpret_cast<const v2f*>(srow + d0));
            v2f a;
            a.x = qon ? __expf((tv.x - cv.x) * invT - tm) * tz : 0.0f;
            a.y = qon ? __expf((tv.y - cv.y) * invT - tm) * tz : 0.0f;
            acc = __builtin_amdgcn_wmma_f32_16x16x4_f32(
                /*neg_a=*/false, a, /*neg_b=*/false, sv,
                /*c_mod=*/(short)0, acc, /*reuse_a=*/false, /*reuse_b=*/false);
        }
    }

    // D[0,n] = acc[0] on lanes 0..15 (n=lane); D[1,n] = acc[1]
    if (lane < NCROPS) {
        atomicAdd(&pdots[lane],      acc[0]);
        atomicAdd(&pdots[16 + lane], acc[1]);
    }
}

// ------------------------------------------------- center EMA update
// new_center[c] = 0.9*center[c] + 0.1 * (sum_{b: tgt[b]==c} t1[b]+t2[b]) / max(2*cnt,1)
__global__ void k_center(const float* __restrict__ t,
                         const float* __restrict__ center,
                         const int*   __restrict__ targets,
                         float* __restrict__ out_center) {
    __shared__ int tg[BATCH];
    const int TILE  = 1024;                   // 256 threads * float4
    const int c     = blockIdx.x / (OUT_DIM / TILE);
    const int tile  = blockIdx.x % (OUT_DIM / TILE);
    if (threadIdx.x < BATCH) tg[threadIdx.x] = targets[threadIdx.x];
    __syncthreads();

    const int d = tile * TILE + threadIdx.x * 4;
    float4 acc = make_float4(0.f, 0.f, 0.f, 0.f);
    int cnt = 0;
    for (int b = 0; b < BATCH; ++b) {
        if (tg[b] == c) {
            ++cnt;
            float4 a1 = *reinterpret_cast<const float4*>(t + (size_t)b * OUT_DIM + d);
            float4 a2 = *reinterpret_cast<const float4*>(t + (size_t)(BATCH + b) * OUT_DIM + d);
            acc.x += a1.x + a2.x;  acc.y += a1.y + a2.y;
            acc.z += a1.z + a2.z;  acc.w += a1.w + a2.w;
        }
    }
    const int cnts = (2 * cnt > 1) ? 2 * cnt : 1;
    const float inv = 0.1f / (float)cnts;
    float4 cv = *reinterpret_cast<const float4*>(center + (size_t)c * OUT_DIM + d);
    float4 o;
    o.x = cv.x * 0.9f + acc.x * inv;
    o.y = cv.y * 0.9f + acc.y * inv;
    o.z = cv.z * 0.9f + acc.z * inv;
    o.w = cv.w * 0.9f + acc.w * inv;
    *reinterpret_cast<float4*>(out_center + (size_t)c * OUT_DIM + d) = o;
}

// ------------------------------------------------- finalize loss scalar
__global__ void k_finalize(const float* __restrict__ pdots,
                           const float* __restrict__ Lsum,
                           float* __restrict__ out) {
    if (threadIdx.x == 0 && blockIdx.x == 0) {
        float total = 0.0f;
        for (int i = 0; i < 2; ++i)
            for (int v = 0; v < NCROPS; ++v)
                if (v != i)
                    total += -(10.0f * pdots[i * 16 + v] - Lsum[v]) * (1.0f / (float)BATCH);
        out[0] = total / (float)(2 * NCROPS - 2);
    }
}

extern "C" void kernel_launch(void* const* d_in, const int* in_sizes, int n_in,
                              void* d_out, int out_size, void* d_ws, size_t ws_size,
                              hipStream_t stream) {
    const float* s  = (const float*)d_in[0];   // student (2560, 65536)
    const float* t  = (const float*)d_in[1];   // teacher (512, 65536)
    const float* c  = (const float*)d_in[2];   // center  (51, 65536)
    const int*   tg = (const int*)  d_in[3];   // targets (256,)
    const int*   ep = (const int*)  d_in[4];   // epoch scalar

    float* out   = (float*)d_out;              // [0]=loss, [1..]=new_center
    float* ws    = (float*)d_ws;
    float* pdots = ws;                         // 32 floats
    float* Lsum  = ws + 32;                    // 16 floats
    float* tmax  = ws + 64;                    // 512 floats
    float* tinvz = ws + 576;                   // 512 floats

    k_init<<<1, 64, 0, stream>>>(ws);
    k_teacher_stats<<<2 * BATCH, 256, 0, stream>>>(t, c, tg, ep, tmax, tinvz);
    k_student_lse<<<NCROPS * BATCH, 256, 0, stream>>>(s, Lsum);
    k_dots_wmma<<<BATCH * 16, 128, 0, stream>>>(s, t, c, tg, ep, tmax, tinvz, pdots);
    k_center<<<N_CENTERS * (OUT_DIM / 1024), 256, 0, stream>>>(t, c, tg, out + 1);
    k_finalize<<<1, 32, 0, stream>>>(pdots, Lsum, out);
}